// GAT_81819126988854
// MI455X (gfx1250) — compile-verified
//
#include <hip/hip_runtime.h>

#define HEADS 4
#define NEG_SLOPE 0.2f

typedef float v2f __attribute__((ext_vector_type(2)));
typedef float v8f __attribute__((ext_vector_type(8)));

#if __has_builtin(__builtin_amdgcn_wmma_f32_16x16x4_f32)
#define USE_WMMA_F32 1
#else
#define USE_WMMA_F32 0
#endif

// ---------------------------- fills / copies --------------------------------
__global__ void fill_i32(int* __restrict__ p, int v, long long n) {
  for (long long i = blockIdx.x * (long long)blockDim.x + threadIdx.x; i < n;
       i += (long long)gridDim.x * blockDim.x)
    p[i] = v;
}
__global__ void copy_i32(int* __restrict__ d, const int* __restrict__ s, long long n) {
  for (long long i = blockIdx.x * (long long)blockDim.x + threadIdx.x; i < n;
       i += (long long)gridDim.x * blockDim.x)
    d[i] = s[i];
}

// ------------------------- CSR build (by destination) -----------------------
__global__ void count_deg(const int* __restrict__ dst, int* __restrict__ deg,
                          long long E) {
  for (long long e = blockIdx.x * (long long)blockDim.x + threadIdx.x; e < E;
       e += (long long)gridDim.x * blockDim.x)
    atomicAdd(&deg[dst[e]], 1);
}

// block-local exclusive scan of deg -> row_ptr, block totals -> bsum
__global__ void scan_local(const int* __restrict__ deg, int* __restrict__ row_ptr,
                           int* __restrict__ bsum, int N) {
  __shared__ int sm[256];
  const int t = threadIdx.x;
  const int i = blockIdx.x * 256 + t;
  const int v = (i < N) ? deg[i] : 0;
  sm[t] = v;
  __syncthreads();
  for (int o = 1; o < 256; o <<= 1) {
    int x = (t >= o) ? sm[t - o] : 0;
    __syncthreads();
    sm[t] += x;
    __syncthreads();
  }
  if (i < N) row_ptr[i] = sm[t] - v;  // exclusive
  if (t == 255) bsum[blockIdx.x] = sm[255];
}

// single-block exclusive scan of block sums (nb <= 256)
__global__ void scan_aux(int* __restrict__ bsum, int nb) {
  __shared__ int sm[256];
  const int t = threadIdx.x;
  const int v = (t < nb) ? bsum[t] : 0;
  sm[t] = v;
  __syncthreads();
  for (int o = 1; o < 256; o <<= 1) {
    int x = (t >= o) ? sm[t - o] : 0;
    __syncthreads();
    sm[t] += x;
    __syncthreads();
  }
  if (t < nb) bsum[t] = sm[t] - v;
}

__global__ void scan_add(int* __restrict__ row_ptr, const int* __restrict__ bsum,
                         int N, int E) {
  const int i = blockIdx.x * 256 + threadIdx.x;
  if (i < N) row_ptr[i] += bsum[blockIdx.x];
  if (i == 0) row_ptr[N] = E;
}

__global__ void fill_eidx(const int* __restrict__ dst, int* __restrict__ cursor,
                          int* __restrict__ eidx, long long E) {
  for (long long e = blockIdx.x * (long long)blockDim.x + threadIdx.x; e < E;
       e += (long long)gridDim.x * blockDim.x) {
    int pos = atomicAdd(&cursor[dst[e]], 1);
    eidx[pos] = (int)e;
  }
}

// ------------------- fp32 WMMA GEMM: C[M,Ncol] = A[M,K] @ B[K,Ncol] (+bias) --
// One wave per 16x16 output tile; K-loop in steps of 4 using
// V_WMMA_F32_16X16X4_F32. The block's K x 16 B-tile is staged in LDS once
// (<= 8 KB of the 320 KB/WGP) so the k-loop reads B via ds_load and only
// streams A from global. Operand layouts per ISA 7.12.2.
__global__ void gemm16_wmma(const float* __restrict__ A, const float* __restrict__ B,
                            float* __restrict__ C, const float* __restrict__ bias,
                            int M, int K, int Ncol) {
  __shared__ float Bs[2048];  // up to K=128 x 16 cols
  const int lane  = threadIdx.x & 31;
  const int wave  = threadIdx.x >> 5;
  const int mtile = blockIdx.x * (blockDim.x >> 5) + wave;
  const int col0  = blockIdx.y * 16;

  for (int t = threadIdx.x; t < K * 16; t += blockDim.x) {
    const int kk = t >> 4, cc = t & 15;
    Bs[t] = B[(long long)kk * Ncol + col0 + cc];
  }
  __syncthreads();

  if (mtile * 16 < M) {
    const int row0 = mtile * 16;
    const int half = lane >> 4;
    const int l15  = lane & 15;
    float out[8];
#if USE_WMMA_F32
    const float* Ar = A + (long long)(row0 + l15) * K;
    v8f acc = {};
    for (int k = 0; k < K; k += 4) {
      const int ka = k + half * 2;
      v2f a, b;
      a.x = Ar[ka];
      a.y = Ar[ka + 1];
      b.x = Bs[ka * 16 + l15];
      b.y = Bs[(ka + 1) * 16 + l15];
      acc = __builtin_amdgcn_wmma_f32_16x16x4_f32(false, a, false, b, (short)0, acc,
                                                  false, false);
    }
#pragma unroll
    for (int i = 0; i < 8; ++i) out[i] = acc[i];
#else
#pragma unroll
    for (int i = 0; i < 8; ++i) out[i] = 0.f;
    for (int k = 0; k < K; ++k) {
      float bv = Bs[k * 16 + l15];
#pragma unroll
      for (int i = 0; i < 8; ++i)
        out[i] += A[(long long)(row0 + half * 8 + i) * K + k] * bv;
    }
#endif
    const int colg = col0 + l15;
    const float bv = bias ? bias[colg] : 0.f;
#pragma unroll
    for (int i = 0; i < 8; ++i)
      C[(long long)(row0 + half * 8 + i) * Ncol + colg] = out[i] + bv;
  }
}

// -------------- per-node attention coefficients aS/aD = <h[n,h,:], a> -------
__global__ void node_alpha(const float* __restrict__ hF, const float* __restrict__ asrc,
                           const float* __restrict__ adst, float* __restrict__ aS,
                           float* __restrict__ aD, long long NH, int C) {
  for (long long i = blockIdx.x * (long long)blockDim.x + threadIdx.x; i < NH;
       i += (long long)gridDim.x * blockDim.x) {
    long long n = i >> 2;
    int h = (int)(i & 3);
    const float* hp = hF + (n * HEADS + h) * C;
    const float* ps = asrc + h * C;
    const float* pd = adst + h * C;
    float s = 0.f, d = 0.f;
    for (int c = 0; c < C; ++c) {
      float v = hp[c];
      s += v * ps[c];
      d += v * pd[c];
    }
    aS[i] = s;
    aD[i] = d;
  }
}

// ---- per (node, head): segment softmax over incident edges, no atomics -----
// Pass A: register max over incident-edge logits. Pass B: w = exp(l - m),
// stored at the CSR slot; sinv = 1/(sum + eps). Empty segments -> out row = 0,
// matching the reference's isfinite guard semantics.
__global__ void node_softmax(const int* __restrict__ rp, const int* __restrict__ eidx,
                             const int* __restrict__ src, const float* __restrict__ aS,
                             const float* __restrict__ aD, float* __restrict__ ew,
                             float* __restrict__ sinv, long long NH) {
  for (long long i = blockIdx.x * (long long)blockDim.x + threadIdx.x; i < NH;
       i += (long long)gridDim.x * blockDim.x) {
    const long long n = i >> 2;
    const int h = (int)(i & 3);
    const int beg = rp[n], end = rp[n + 1];
    const float ad = aD[i];
    float m = -3.4e38f;
    for (int j = beg; j < end; ++j) {
      const int e = eidx[j];
      float l = aS[(long long)src[e] * HEADS + h] + ad;
      l = l > 0.f ? l : NEG_SLOPE * l;
      m = l > m ? l : m;
    }
    float s = 0.f;
    for (int j = beg; j < end; ++j) {
      const int e = eidx[j];
      float l = aS[(long long)src[e] * HEADS + h] + ad;
      l = l > 0.f ? l : NEG_SLOPE * l;
      const float w = __expf(l - m);
      ew[(long long)j * HEADS + h] = w;
      s += w;
    }
    sinv[i] = 1.f / (s + 1e-16f);
  }
}

// ---- per (node, head): gather-reduce of h[src]*alpha, plain stores ---------
template <int C>
__global__ void node_aggr(const int* __restrict__ rp, const int* __restrict__ eidx,
                          const int* __restrict__ src, const float* __restrict__ ew,
                          const float* __restrict__ sinv, const float* __restrict__ hF,
                          float* __restrict__ out, long long NH) {
  for (long long i = blockIdx.x * (long long)blockDim.x + threadIdx.x; i < NH;
       i += (long long)gridDim.x * blockDim.x) {
    const long long n = i >> 2;
    const int h = (int)(i & 3);
    const int beg = rp[n], end = rp[n + 1];
    float acc[C];
#pragma unroll
    for (int c = 0; c < C; ++c) acc[c] = 0.f;
    for (int j = beg; j < end; ++j) {
      const int e = eidx[j];
      const long long s = src[e];
      const float w = ew[(long long)j * HEADS + h];
      const float* __restrict__ hr = hF + (s * HEADS + h) * C;
#pragma unroll
      for (int c = 0; c < C; ++c) acc[c] += hr[c] * w;
    }
    const float si = sinv[i];
    float* __restrict__ op = out + (n * HEADS + h) * C;
#pragma unroll
    for (int c = 0; c < C; ++c) op[c] = acc[c] * si;
  }
}

// ---- layer-1 finalize: x1 = elu(acc + b1) in place -------------------------
__global__ void finalize_elu(float* __restrict__ acc, const float* __restrict__ bias,
                             long long n, int F) {
  for (long long i = blockIdx.x * (long long)blockDim.x + threadIdx.x; i < n;
       i += (long long)gridDim.x * blockDim.x) {
    float v = acc[i] + bias[(int)(i % F)];
    acc[i] = v > 0.f ? v : (__expf(v) - 1.f);
  }
}

// ---- layer-2 finalize: mean over heads + b2, then log_softmax --------------
__global__ void finalize_out(const float* __restrict__ acc, const float* __restrict__ b2,
                             float* __restrict__ out, int N) {
  for (int n = blockIdx.x * blockDim.x + threadIdx.x; n < N;
       n += gridDim.x * blockDim.x) {
    const float* p = acc + (long long)n * 64;
    float v[16];
    float mx = -3.4e38f;
#pragma unroll
    for (int c = 0; c < 16; ++c) {
      float t = 0.25f * (p[c] + p[16 + c] + p[32 + c] + p[48 + c]) + b2[c];
      v[c] = t;
      mx = t > mx ? t : mx;
    }
    float s = 0.f;
#pragma unroll
    for (int c = 0; c < 16; ++c) s += __expf(v[c] - mx);
    float lse = mx + __logf(s);
#pragma unroll
    for (int c = 0; c < 16; ++c) out[(long long)n * 16 + c] = v[c] - lse;
  }
}

extern "C" void kernel_launch(void* const* d_in, const int* in_sizes, int n_in,
                              void* d_out, int out_size, void* d_ws, size_t ws_size,
                              hipStream_t stream) {
  const float* x    = (const float*)d_in[0];
  const int*   ei   = (const int*)d_in[1];
  const float* Wemb = (const float*)d_in[2];
  const float* bemb = (const float*)d_in[3];
  const float* W1   = (const float*)d_in[4];
  const float* as1  = (const float*)d_in[5];
  const float* ad1  = (const float*)d_in[6];
  const float* b1   = (const float*)d_in[7];
  const float* W2   = (const float*)d_in[8];
  const float* as2  = (const float*)d_in[9];
  const float* ad2  = (const float*)d_in[10];
  const float* b2   = (const float*)d_in[11];

  const int FIN = 128, HID = 32, OUT = 16;
  const long long N = in_sizes[0] / FIN;
  const long long E = in_sizes[1] / 2;
  const int* src = ei;
  const int* dst = ei + E;

  char* ws = (char*)d_ws;
  size_t off = 0;
  auto alloc = [&](size_t bytes) -> void* {
    void* p = ws + off;
    off += (bytes + 255) & ~(size_t)255;
    return p;
  };
  // graph structure (built once, shared by both layers)
  int*   row_ptr = (int*)alloc((size_t)(N + 1) * 4);
  int*   deg     = (int*)alloc((size_t)N * 4);  // reused as fill cursor
  int*   bsum    = (int*)alloc(256 * 4);
  int*   eidx    = (int*)alloc((size_t)E * 4);
  // features / per-layer scratch
  float* h0   = (float*)alloc((size_t)N * HID * 4);          // x@Wemb+bemb
  float* hA   = (float*)alloc((size_t)N * HEADS * HID * 4);  // layer1 feats; reused as acc2
  float* aS   = (float*)alloc((size_t)N * HEADS * 4);
  float* aD   = (float*)alloc((size_t)N * HEADS * 4);
  float* sinv = (float*)alloc((size_t)N * HEADS * 4);
  float* ew   = (float*)alloc((size_t)E * HEADS * 4);
  float* acc1 = (float*)alloc((size_t)N * HEADS * HID * 4);  // becomes x1 after ELU
  float* hB   = (float*)alloc((size_t)N * HEADS * OUT * 4);  // layer2 feats
  float* acc2 = hA;                                          // hA dead after layer1

  const dim3 blk(256);
  const long long NH = N * HEADS;
  auto cdiv = [](long long a, long long b) { return (unsigned)((a + b - 1) / b); };
  const int mt = (int)((N + 15) / 16);
  const int nb = (int)((N + 255) / 256);  // 196 block sums (fits 1-block aux scan)

  // ---------- CSR build (dst-major), shared by both layers ----------
  fill_i32<<<cdiv(N, 256), blk, 0, stream>>>(deg, 0, N);
  count_deg<<<cdiv(E, 256), blk, 0, stream>>>(dst, deg, E);
  scan_local<<<nb, blk, 0, stream>>>(deg, row_ptr, bsum, (int)N);
  scan_aux<<<1, blk, 0, stream>>>(bsum, nb);
  scan_add<<<nb, blk, 0, stream>>>(row_ptr, bsum, (int)N, (int)E);
  copy_i32<<<cdiv(N, 256), blk, 0, stream>>>(deg, row_ptr, N);  // deg -> cursor
  fill_eidx<<<cdiv(E, 256), blk, 0, stream>>>(dst, deg, eidx, E);

  // ---------- embedding GEMM: h0 = x @ Wemb + bemb ----------
  gemm16_wmma<<<dim3((mt + 7) / 8, HID / 16), blk, 0, stream>>>(
      x, Wemb, h0, bemb, (int)N, FIN, HID);

  // ---------- GAT layer 1 ----------
  gemm16_wmma<<<dim3((mt + 7) / 8, (HEADS * HID) / 16), blk, 0, stream>>>(
      h0, W1, hA, nullptr, (int)N, HID, HEADS * HID);
  node_alpha<<<cdiv(NH, 256), blk, 0, stream>>>(hA, as1, ad1, aS, aD, NH, HID);
  node_softmax<<<cdiv(NH, 256), blk, 0, stream>>>(row_ptr, eidx, src, aS, aD, ew,
                                                  sinv, NH);
  node_aggr<32><<<cdiv(NH, 256), blk, 0, stream>>>(row_ptr, eidx, src, ew, sinv,
                                                   hA, acc1, NH);
  finalize_elu<<<cdiv(N * HEADS * HID, 256), blk, 0, stream>>>(
      acc1, b1, N * HEADS * HID, HEADS * HID);

  // ---------- GAT layer 2 ----------
  gemm16_wmma<<<dim3((mt + 7) / 8, (HEADS * OUT) / 16), blk, 0, stream>>>(
      acc1, W2, hB, nullptr, (int)N, HEADS * HID, HEADS * OUT);
  node_alpha<<<cdiv(NH, 256), blk, 0, stream>>>(hB, as2, ad2, aS, aD, NH, OUT);
  node_softmax<<<cdiv(NH, 256), blk, 0, stream>>>(row_ptr, eidx, src, aS, aD, ew,
                                                  sinv, NH);
  node_aggr<16><<<cdiv(NH, 256), blk, 0, stream>>>(row_ptr, eidx, src, ew, sinv,
                                                   hB, acc2, NH);
  finalize_out<<<cdiv(N, 128), dim3(128), 0, stream>>>(acc2, b2, (float*)d_out,
                                                       (int)N);
}